// AInnoFaceLoss_40879498728872
// MI455X (gfx1250) — compile-verified
//
#include <hip/hip_runtime.h>
#include <stdint.h>

#define GTN 64  // ground-truth boxes per batch (fixed by reference: G = 64)

// One thread per (batch, anchor). GT boxes for the block's batch are staged in
// LDS via the CDNA5 async global->LDS copy path, converted once to
// (x1,y1,x2,y2) + area, then each thread runs the 64-way argmax with
// division-free IoU comparison (cross-multiplication; unions are > 0).
__global__ __launch_bounds__(256) void iou_assign_kernel(
    const float* __restrict__ anchors,   // (A,4) xywh
    const float* __restrict__ gt,        // (B,G,4) xywh
    float*  __restrict__ out_score,      // (B,A)
    float4* __restrict__ out_bbox,       // (B,A,4) xyxy of best GT
    int A)
{
    __shared__ float  rawS[GTN * 4];   // async-copied raw xywh
    __shared__ float4 boxS[GTN];       // converted xyxy
    __shared__ float  areaS[GTN];      // w*h per GT

    const int tid = threadIdx.x;
    const int b   = blockIdx.y;
    const int a   = blockIdx.x * blockDim.x + tid;

    // ---- Stage GT tile into LDS with gfx1250 async copy (ASYNCcnt path) ----
    // tid<64 spans waves 0..1 wholly (wave32), so the asm below is wave-uniform.
    if (tid < GTN) {
        const uint64_t src = (uint64_t)(uintptr_t)(gt + ((size_t)b * GTN + tid) * 4);
        // Low 32 bits of a generic pointer into LDS == wave-relative LDS byte
        // offset (ISA 10.2 aperture mapping) == what VDST of the async op wants.
        const uint32_t dst = (uint32_t)(uintptr_t)(&rawS[tid * 4]);
        asm volatile("global_load_async_to_lds_b128 %0, %1, off"
                     :
                     : "v"(dst), "v"(src)
                     : "memory");
        asm volatile("s_wait_asynccnt 0" ::: "memory");
        const float gx = rawS[tid * 4 + 0];
        const float gy = rawS[tid * 4 + 1];
        const float gw = rawS[tid * 4 + 2];
        const float gh = rawS[tid * 4 + 3];
        boxS[tid]  = make_float4(gx, gy, gx + gw, gy + gh);
        areaS[tid] = gw * gh;
    }
    __syncthreads();

    if (a >= A) return;

    // Anchor box (coalesced b128 load), convert to xyxy + area once.
    const float4 av  = *(const float4*)(anchors + (size_t)a * 4);
    const float ax1 = av.x,        ay1 = av.y;
    const float ax2 = av.x + av.z, ay2 = av.y + av.w;
    const float areaA = av.z * av.w;

    // Division-free running argmax: iou_g > iou_best  <=>
    // inter_g * uni_best > inter_best * uni_g   (all unions > 0).
    // Init (bInter=-1, bUni=1) guarantees g=0 wins the first compare, and the
    // strict '>' afterwards reproduces jnp.argmax first-max tie-breaking.
    float bInter = -1.0f;
    float bUni   =  1.0f;
    int   bestIdx = 0;

    #pragma unroll 16
    for (int g = 0; g < GTN; ++g) {
        const float4 gb = boxS[g];          // broadcast ds_load_b128
        const float x1 = fmaxf(ax1, gb.x);
        const float y1 = fmaxf(ay1, gb.y);
        const float x2 = fminf(ax2, gb.z);
        const float y2 = fminf(ay2, gb.w);
        const float w  = fmaxf(x2 - x1, 0.0f);
        const float h  = fmaxf(y2 - y1, 0.0f);
        const float inter = w * h;
        const float uni   = areaA + areaS[g] - inter;
        const bool better = inter * bUni > bInter * uni;
        bInter  = better ? inter : bInter;
        bUni    = better ? uni   : bUni;
        bestIdx = better ? g     : bestIdx;
    }

    const size_t o = (size_t)b * A + a;
    out_score[o] = bInter / bUni;   // the single division per anchor
    out_bbox[o]  = boxS[bestIdx];   // xyxy of best GT, b128 store
}

extern "C" void kernel_launch(void* const* d_in, const int* in_sizes, int n_in,
                              void* d_out, int out_size, void* d_ws, size_t ws_size,
                              hipStream_t stream) {
    // setup_inputs order: fs_proposal, ss_proposal, anchors, ground_truth.
    // fs/ss are unused by the reference.
    const float* anchors = (const float*)d_in[2];
    const float* gt      = (const float*)d_in[3];
    const int A = in_sizes[2] / 4;          // 100000
    const int B = in_sizes[3] / (GTN * 4);  // 4

    // d_out layout: target_score (B*A) then target_bbox (B*A*4), both fp32.
    float*  out_score = (float*)d_out;
    float4* out_bbox  = (float4*)((float*)d_out + (size_t)B * A);

    dim3 block(256, 1, 1);
    dim3 grid((A + 255) / 256, B, 1);
    iou_assign_kernel<<<grid, block, 0, stream>>>(anchors, gt, out_score, out_bbox, A);
}